// Sparsegen_linear2_472446402760
// MI455X (gfx1250) — compile-verified
//
#include <hip/hip_runtime.h>
#include <stdint.h>

// sparsegen-lin (lambda = 0.5): out = sparsemax(2*x) per row of 2048 f32.
// One row per wave32; async global->LDS staging; sort-free candidate-pruned
// Newton (Michelot) for the simplex threshold tau.

typedef float v4f __attribute__((ext_vector_type(4)));
typedef int   v4i __attribute__((vector_size(4 * sizeof(int))));

#define N_COLS          2048
#define WAVES_PER_BLOCK 8
#define BLOCK_THREADS   (WAVES_PER_BLOCK * 32)
#define CHUNKS_PER_LANE 16              // 16 chunks * 32 lanes * 4 f32 = 2048
#define CAND_CAP        32              // candidate set held one-per-lane
#define NEWTON_ITERS    24
#define FALLBACK_ITERS  48

// ---- CDNA5 async global->LDS copy (ASYNCcnt-tracked) -----------------------
// Builtin signature (from hipcc diagnostic): arg0 = AS1 (global) v4i*,
// arg1 = AS3 (LDS) v4i*, then imm offset, imm cpol.
#if __has_builtin(__builtin_amdgcn_global_load_async_to_lds_b128)
#define HAVE_ASYNC_LDS 1
#define ASYNC_COPY_B128(gsrc, ldst)                                           \
  __builtin_amdgcn_global_load_async_to_lds_b128(                             \
      (__attribute__((address_space(1))) v4i*)(v4i*)(float*)(gsrc),           \
      (__attribute__((address_space(3))) v4i*)(v4i*)(float*)(ldst), 0, 0)
#else
#define HAVE_ASYNC_LDS 0
#define ASYNC_COPY_B128(gsrc, ldst)                                           \
  (*(v4f*)(ldst) = *(const v4f*)(gsrc))
#endif

static __device__ __forceinline__ void wait_async_all() {
#if HAVE_ASYNC_LDS
#if __has_builtin(__builtin_amdgcn_s_wait_asynccnt)
  __builtin_amdgcn_s_wait_asynccnt(0);
#else
  asm volatile("s_wait_asynccnt 0" ::: "memory");
#endif
#endif
}

// ---- wave32 reductions -----------------------------------------------------
static __device__ __forceinline__ float wave_max(float v) {
#pragma unroll
  for (int off = 16; off > 0; off >>= 1)
    v = fmaxf(v, __shfl_xor(v, off, 32));
  return v;
}

static __device__ __forceinline__ float wave_sum(float v) {
#pragma unroll
  for (int off = 16; off > 0; off >>= 1)
    v += __shfl_xor(v, off, 32);
  return v;
}

__global__ __launch_bounds__(BLOCK_THREADS, 1) void sparsegen_rows_kernel(
    const float* __restrict__ x, float* __restrict__ out, int rows) {
  __shared__ float srow[WAVES_PER_BLOCK][N_COLS];     // 64 KB staging
  __shared__ float scand[WAVES_PER_BLOCK][CAND_CAP];  // support candidates (z)
  __shared__ int   scnt[WAVES_PER_BLOCK];

  const int wave = threadIdx.x >> 5;
  const int lane = threadIdx.x & 31;
  const int row  = blockIdx.x * WAVES_PER_BLOCK + wave;
  if (row >= rows) return;  // wave-uniform

  const float* rp = x + (size_t)row * N_COLS;
  float*       op = out + (size_t)row * N_COLS;
  float*       lr = &srow[wave][0];

  // ---- stage the row into LDS: 16 async b128 copies per lane (coalesced:
  // each instruction moves 32 lanes * 16B = 512B contiguous) ----
#pragma unroll
  for (int j = 0; j < CHUNKS_PER_LANE; ++j) {
    const int e = (j * 32 + lane) * 4;
    ASYNC_COPY_B128(rp + e, lr + e);
  }
  if (lane == 0) scnt[wave] = 0;
  wait_async_all();

  // ---- pass A: row max (z = 2x, exact scaling; max_z = 2*max_x) ----
  float m = -__builtin_inff();
#pragma unroll
  for (int j = 0; j < CHUNKS_PER_LANE; ++j) {
    v4f v = *(const v4f*)(lr + (j * 32 + lane) * 4);
    m = fmaxf(m, fmaxf(fmaxf(v.x, v.y), fmaxf(v.z, v.w)));
  }
  const float xmax = wave_max(m);
  const float zmax = 2.0f * xmax;

  // ---- pass B: compact support candidates. tau* >= zmax-1, so only
  // elements with z > zmax-1  (x > xmax-0.5) can be in the support. ----
  const float thr = xmax - 0.5f;
#pragma unroll
  for (int j = 0; j < CHUNKS_PER_LANE; ++j) {
    v4f v = *(const v4f*)(lr + (j * 32 + lane) * 4);
#pragma unroll
    for (int c = 0; c < 4; ++c) {
      float f = v[c];
      if (f > thr) {
        int idx = atomicAdd(&scnt[wave], 1);
        if (idx < CAND_CAP) scand[wave][idx] = 2.0f * f;
      }
    }
  }
  __threadfence_block();
  const int m_cnt = scnt[wave];

  // ---- Newton (Michelot) from the lower bound tau0 = zmax - 1:
  // f(tau) = sum(max(z - tau, 0)) - 1 is convex piecewise-linear decreasing;
  // tangent steps from below are monotone, never overshoot, finite converge.
  float tau = zmax - 1.0f;
  if (m_cnt <= CAND_CAP) {
    // Fast path: one candidate per lane, pure register/crosslane iterations.
    float cz = (lane < m_cnt) ? scand[wave][lane] : -__builtin_inff();
    for (int it = 0; it < NEWTON_ITERS; ++it) {
      float d = cz - tau;
      float p = d > 0.0f ? d : 0.0f;
      float q = d > 0.0f ? 1.0f : 0.0f;
      float s = wave_sum(p);
      float c = wave_sum(q);
      tau += (s - 1.0f) / c;  // c >= 1: the max element always satisfies d>0
    }
  } else {
    // Rare fallback (dense support): Newton over the full LDS-resident row.
    for (int it = 0; it < FALLBACK_ITERS; ++it) {
      float sl = 0.0f, cl = 0.0f;
      for (int j = 0; j < CHUNKS_PER_LANE; ++j) {
        v4f v = *(const v4f*)(lr + (j * 32 + lane) * 4);
#pragma unroll
        for (int c = 0; c < 4; ++c) {
          float d = fmaf(v[c], 2.0f, -tau);
          if (d > 0.0f) { sl += d; cl += 1.0f; }
        }
      }
      float s = wave_sum(sl);
      float c = wave_sum(cl);
      tau += (s - 1.0f) / c;
    }
  }

  // ---- pass C: p = max(2x - tau, 0), nontemporal streaming stores ----
#pragma unroll
  for (int j = 0; j < CHUNKS_PER_LANE; ++j) {
    const int e = (j * 32 + lane) * 4;
    v4f v = *(const v4f*)(lr + e);
    v4f o;
    o.x = fmaxf(fmaf(v.x, 2.0f, -tau), 0.0f);
    o.y = fmaxf(fmaf(v.y, 2.0f, -tau), 0.0f);
    o.z = fmaxf(fmaf(v.z, 2.0f, -tau), 0.0f);
    o.w = fmaxf(fmaf(v.w, 2.0f, -tau), 0.0f);
    __builtin_nontemporal_store(o, (v4f*)(op + e));
  }
}

extern "C" void kernel_launch(void* const* d_in, const int* in_sizes, int n_in,
                              void* d_out, int out_size, void* d_ws,
                              size_t ws_size, hipStream_t stream) {
  (void)n_in; (void)out_size; (void)d_ws; (void)ws_size;
  const float* x  = (const float*)d_in[0];
  float*       out = (float*)d_out;
  const int rows = in_sizes[0] / N_COLS;  // 65536
  const int grid = (rows + WAVES_PER_BLOCK - 1) / WAVES_PER_BLOCK;
  sparsegen_rows_kernel<<<grid, BLOCK_THREADS, 0, stream>>>(x, out, rows);
}